// BlocksparseMatmulSSS_67259187855493
// MI455X (gfx1250) — compile-verified
//
#include <hip/hip_runtime.h>
#include <hip/hip_bf16.h>
#include <stdint.h>

typedef __attribute__((ext_vector_type(2))) float v2f;
typedef __attribute__((ext_vector_type(8))) float v8f;
typedef int v4i __attribute__((vector_size(16)));

#define BSZ      128   // sparsity block size
#define KSLAB    16    // K-slab staged in LDS per pipeline stage
#define A_STRIDE 20    // LDS row stride (floats) for A [128][20]   -> conflict-free
#define BP_STRIDE 288  // LDS row stride (floats) for B pair rows   -> 288 % 64 == 32

// ---------------- CDNA5 async global->LDS plumbing -------------------------
// Signatures per hipcc diagnostics:
//   b128: (v4i __device__*,  <lds ptr>, Imm offset, Imm cpol)
//   b32 : (int __device__*,  <lds ptr>, Imm offset, Imm cpol)
#if __has_builtin(__builtin_amdgcn_global_load_async_to_lds_b128) && \
    __has_builtin(__builtin_amdgcn_global_load_async_to_lds_b32)
#define HAVE_ASYNC_LDS 1
typedef __attribute__((address_space(1))) v4i* g128_t;
typedef __attribute__((address_space(3))) v4i* l128_t;
typedef __attribute__((address_space(1))) int* g32_t;
typedef __attribute__((address_space(3))) int* l32_t;
__device__ __forceinline__ void async_b128(const void* g, void* l) {
  __builtin_amdgcn_global_load_async_to_lds_b128(
      (g128_t)(uintptr_t)g, (l128_t)(unsigned)(uintptr_t)l, 0, 0);
}
__device__ __forceinline__ void async_b32(const void* g, void* l) {
  __builtin_amdgcn_global_load_async_to_lds_b32(
      (g32_t)(uintptr_t)g, (l32_t)(unsigned)(uintptr_t)l, 0, 0);
}
#else
#define HAVE_ASYNC_LDS 0
#endif

__device__ __forceinline__ void wait_async0() {
#if __has_builtin(__builtin_amdgcn_s_wait_asynccnt)
  __builtin_amdgcn_s_wait_asynccnt(0);
#else
  asm volatile("s_wait_asynccnt 0x0" ::: "memory");
#endif
}

// ---------------------------------------------------------------------------
// Reverse-LUT builder: rev[e] = (global prefix count of ones up to e) - 1 if
// layout[e]!=0 else -1.  Matches the reference's cumsum over the flat array.
// ---------------------------------------------------------------------------
__global__ void build_rev_lut(const int* __restrict__ layout,
                              int* __restrict__ rev, int total) {
  __shared__ int chunkSum[64];
  int t = threadIdx.x;
  if (t < 64) {
    int s = 0;
    int base = t << 7;
    int end = base + 128; if (end > total) end = total;
    for (int f = base; f < end; ++f) s += (layout[f] != 0);
    chunkSum[t] = s;
  }
  __syncthreads();
  int e = blockIdx.x * blockDim.x + t;
  if (e >= total) return;
  int g = e >> 7;
  int cnt = 0;
  for (int q = 0; q < g; ++q) cnt += chunkSum[q];
  for (int f = (g << 7); f <= e; ++f) cnt += (layout[f] != 0);
  rev[e] = (layout[e] != 0) ? (cnt - 1) : -1;
}

// Output LUT: olut[slot] = flat index e of the slot-th nonzero (row-major).
__global__ void build_out_lut(const int* __restrict__ layout,
                              int* __restrict__ olut, int total) {
  __shared__ int chunkSum[64];
  int t = threadIdx.x;
  if (t < 64) {
    int s = 0;
    int base = t << 7;
    int end = base + 128; if (end > total) end = total;
    for (int f = base; f < end; ++f) s += (layout[f] != 0);
    chunkSum[t] = s;
  }
  __syncthreads();
  int e = blockIdx.x * blockDim.x + t;
  if (e >= total) return;
  if (layout[e] == 0) return;
  int g = e >> 7;
  int cnt = 0;
  for (int q = 0; q < g; ++q) cnt += chunkSum[q];
  for (int f = (g << 7); f <= e; ++f) cnt += (layout[f] != 0);
  olut[cnt - 1] = e;
}

// ---------------------------------------------------------------------------
// Main block-sparse GEMM: one workgroup (8 wave32) per 128x128 output block.
// Wave grid 4x2; each wave computes a 32x64 region = 2x4 tiles of 16x16 via
// v_wmma_f32_16x16x4_f32.  K staged through double-buffered LDS slabs of 16
// filled by GLOBAL_LOAD_ASYNC_TO_LDS (no VGPR staging, no spills).  B is
// stored K-pair interleaved so each B fragment is ONE aligned ds_load_b64.
// ---------------------------------------------------------------------------
__global__ __launch_bounds__(256)
void bsmm_sss_wmma(const float* __restrict__ x, const float* __restrict__ y,
                   const int* __restrict__ revx, const int* __restrict__ revy,
                   const int* __restrict__ olut, float* __restrict__ out,
                   int R, int K, int C) {
  __shared__ float As[2][BSZ * A_STRIDE];             // A slab: [m][k]
  __shared__ float Bs[2][(KSLAB / 2) * BP_STRIDE];    // B slab: [kpair][n*2 + k&1]
  __shared__ int ixListS[32], iyListS[32];
  __shared__ int nValidS;

  const int tid  = threadIdx.x;
  const int lane = tid & 31;
  const int wave = tid >> 5;
  const int rowBase = (wave >> 1) * 32;   // 4 wave-rows
  const int colBase = (wave & 1) * 64;    // 2 wave-cols
  const int mlo = lane & 15;
  const int h   = lane >> 4;              // lane half: K pair select / M+8

  // Decode this block's (b, r, c) from the output LUT.
  const int e = olut[blockIdx.x];
  const int c = e % C;
  const int r = (e / C) % R;
  const int b = e / (C * R);

  // Wave 0 builds the ordered list of valid k-blocks (both operands present).
  if (wave == 0) {
    int k  = lane;                                   // K <= 32
    int ix = (k < K) ? revx[(b * R + r) * K + k] : -1;
    int iy = (k < K) ? revy[(b * K + k) * C + c] : -1;
    bool valid = (ix >= 0) && (iy >= 0);
    unsigned long long m = __ballot(valid ? 1 : 0);
    int pos = __popcll(m & ((1ull << lane) - 1ull));
    if (valid) { ixListS[pos] = ix; iyListS[pos] = iy; }
    if (lane == 0) nValidS = (int)__popcll(m);
  }
  __syncthreads();
  const int nv  = nValidS;
  const int NIT = nv * (BSZ / KSLAB);     // total slab iterations

  v8f acc[2][4];
#pragma unroll
  for (int tm = 0; tm < 2; ++tm)
#pragma unroll
    for (int tn = 0; tn < 4; ++tn)
#pragma unroll
      for (int v = 0; v < 8; ++v) acc[tm][tn][v] = 0.0f;

#if !HAVE_ASYNC_LDS
  float4 aR[2], bR[2];
#endif

  // Issue the full slab copy for iteration `it` into LDS buffer (it&1).
  auto AGLOAD = [&](int it) {
    int buf = it & 1;
    int kb  = it >> 3;           // which valid k-block
    int kk  = (it & 7) * KSLAB;  // K offset inside the 128-K block
    const float* xb = x + (size_t)ixListS[kb] * (BSZ * BSZ);
    const float* yb = y + (size_t)iyListS[kb] * (BSZ * BSZ);
#pragma unroll
    for (int i = 0; i < 2; ++i) {
      int q  = i * 256 + tid;
      int mA = q >> 2, k4 = (q & 3) << 2;            // A: 128 rows x 16 k
      int kB = q >> 5, n4 = (q & 31) << 2;           // B: 16 k x 128 n
#if HAVE_ASYNC_LDS
      async_b128(xb + mA * BSZ + kk + k4, &As[buf][mA * A_STRIDE + k4]);
      const float* gsrc = yb + (size_t)(kk + kB) * BSZ + n4;
      float* dst = &Bs[buf][(kB >> 1) * BP_STRIDE + (kB & 1)];
#pragma unroll
      for (int j = 0; j < 4; ++j)                    // pair-interleave scatter
        async_b32(gsrc + j, &dst[(n4 + j) * 2]);
#else
      aR[i] = *(const float4*)(xb + mA * BSZ + kk + k4);
      bR[i] = *(const float4*)(yb + (size_t)(kk + kB) * BSZ + n4);
#endif
    }
#if !HAVE_ASYNC_LDS
#pragma unroll
    for (int i = 0; i < 2; ++i) {
      int q  = i * 256 + tid;
      int mA = q >> 2, k4 = (q & 3) << 2;
      *(float4*)&As[buf][mA * A_STRIDE + k4] = aR[i];
      int kB = q >> 5, n4 = (q & 31) << 2;
      const float* src = (const float*)&bR[i];
      float* dst = &Bs[buf][(kB >> 1) * BP_STRIDE + (kB & 1)];
#pragma unroll
      for (int j = 0; j < 4; ++j)
        dst[(n4 + j) * 2] = src[j];
    }
#endif
  };

  if (NIT > 0) AGLOAD(0);
  wait_async0();
  __syncthreads();

  for (int it = 0; it < NIT; ++it) {
    const int  cur  = it & 1;
    const bool more = (it + 1) < NIT;
    if (more) AGLOAD(it + 1);   // async DMA for next slab runs under compute

#pragma unroll
    for (int k0 = 0; k0 < KSLAB; k0 += 4) {
      // A fragment (16x4 f32): lane = 16*h + M, VGPR j -> K = k0 + j + 2*h
      v2f a0 = *(const v2f*)&As[cur][(rowBase +      mlo) * A_STRIDE + k0 + 2 * h];
      v2f a1 = *(const v2f*)&As[cur][(rowBase + 16 + mlo) * A_STRIDE + k0 + 2 * h];
      const int kp = (k0 >> 1) + h;    // B pair row for this lane half
#pragma unroll
      for (int tn = 0; tn < 4; ++tn) {
        int n = colBase + tn * 16 + mlo;
        // B fragment: single aligned b64 read, components are K=k0+2h+{0,1}
        v2f bf = *(const v2f*)&Bs[cur][kp * BP_STRIDE + n * 2];
        acc[0][tn] = __builtin_amdgcn_wmma_f32_16x16x4_f32(
            false, a0, false, bf, (short)0, acc[0][tn], false, false);
        acc[1][tn] = __builtin_amdgcn_wmma_f32_16x16x4_f32(
            false, a1, false, bf, (short)0, acc[1][tn], false, false);
      }
    }
    if (more) { wait_async0(); __syncthreads(); }
  }

  // C/D layout: VGPR v, lane -> row = v + 8*h, col = lane&15 inside the tile.
  float* ob = out + (size_t)blockIdx.x * (BSZ * BSZ);
#pragma unroll
  for (int tm = 0; tm < 2; ++tm)
#pragma unroll
    for (int tn = 0; tn < 4; ++tn)
#pragma unroll
      for (int v = 0; v < 8; ++v) {
        int row = rowBase + tm * 16 + v + 8 * h;
        int col = colBase + tn * 16 + mlo;
        ob[row * BSZ + col] = acc[tm][tn][v];
      }
}

// ---------------------------------------------------------------------------
extern "C" void kernel_launch(void* const* d_in, const int* in_sizes, int n_in,
                              void* d_out, int out_size, void* d_ws, size_t ws_size,
                              hipStream_t stream) {
  const float* x  = (const float*)d_in[0];
  const float* y  = (const float*)d_in[1];
  const int*   lx = (const int*)d_in[2];   // [B,R,K] = [8,32,32]
  const int*   ly = (const int*)d_in[3];   // [B,K,C]
  const int*   lo = (const int*)d_in[4];   // [B,R,C]
  // d_in[5] = o_n_sparse_blocks (device scalar; grid derived from out_size)

  const int R = 32, K = 32, C = 32;
  const int totX = in_sizes[2];
  const int totY = in_sizes[3];
  const int totO = in_sizes[4];
  const int nOut = out_size / (BSZ * BSZ);

  int* revx = (int*)d_ws;
  int* revy = revx + totX;
  int* olut = revy + totY;

  build_rev_lut<<<(totX + 255) / 256, 256, 0, stream>>>(lx, revx, totX);
  build_rev_lut<<<(totY + 255) / 256, 256, 0, stream>>>(ly, revy, totY);
  build_out_lut<<<(totO + 255) / 256, 256, 0, stream>>>(lo, olut, totO);

  bsmm_sss_wmma<<<dim3(nOut), dim3(256), 0, stream>>>(
      x, y, revx, revy, olut, (float*)d_out, R, K, C);
}